// PytorchLlamaDecoderModel_17772574671280
// MI455X (gfx1250) — compile-verified
//
#include <hip/hip_runtime.h>
#include <math.h>

#define DIM  4096
#define NH   32
#define HD   128
#define FFN  11008
#define SPN  2048
#define SEQ  (SPN + 1)   // 2049 keys (cache + new)
#define NB   8
#define VCH  32          // V rows per TDM chunk
#define NCH  (SPN / VCH) // 64 chunks

typedef float v2f __attribute__((ext_vector_type(2)));
typedef float v8f __attribute__((ext_vector_type(8)));
typedef unsigned int u32x4 __attribute__((ext_vector_type(4)));
typedef int i32x4 __attribute__((ext_vector_type(4)));
typedef int i32x8 __attribute__((ext_vector_type(8)));

// ---------------------------------------------------------------------------
// TDM: issue tensor_load_to_lds of a 2D tile: `rows` rows of HD fp32 elements,
// row stride = NH*HD elements, into LDS at lds_off. (D# per ISA 8.3/8.4)
// 6-arg builtin form (amdgpu-toolchain / clang-23).
// ---------------------------------------------------------------------------
__device__ __forceinline__ void tdm_load_v(unsigned int lds_off,
                                           const float* gptr, int rows) {
  unsigned long long ga = (unsigned long long)(uintptr_t)gptr;
  u32x4 g0;
  g0.x = 1u;                                        // count=1, user mode
  g0.y = lds_off;                                   // lds_addr [63:32]
  g0.z = (unsigned int)ga;                          // global_addr lo
  g0.w = (unsigned int)((ga >> 32) & 0x1FFFFFFu)    // global_addr [56:32]
         | (2u << 30);                              // type=2 ("image")
  i32x8 g1;
  g1[0] = 0x00020000;                               // data_size=2 (4 bytes)
  g1[1] = (int)((HD & 0xFFFF) << 16);               // tensor_dim0 lo16 @ [63:48]
  g1[2] = (int)(((HD >> 16) & 0xFFFF)               // tensor_dim0 hi16
                | ((SPN & 0xFFFF) << 16));          // tensor_dim1 lo16
  g1[3] = (int)(((SPN >> 16) & 0xFFFF)              // tensor_dim1 hi16
                | ((HD & 0xFFFF) << 16));           // tile_dim0 @ [127:112]
  g1[4] = rows & 0xFFFF;                            // tile_dim1
  g1[5] = NH * HD;                                  // tensor_dim0_stride lo32
  g1[6] = 0;
  g1[7] = 0;
  i32x4 z4 = {0, 0, 0, 0};
  i32x8 z8 = {0, 0, 0, 0, 0, 0, 0, 0};
  __builtin_amdgcn_tensor_load_to_lds(g0, g1, z4, z4, z8, 0);
}

// ---------------------------------------------------------------------------
// RMSNorm: one block per batch row (8 rows of 4096)
// ---------------------------------------------------------------------------
__global__ __launch_bounds__(256)
void rmsnorm_kernel(const float* __restrict__ x, const float* __restrict__ w,
                    float* __restrict__ out) {
  int row = blockIdx.x;
  const float* xr = x + (size_t)row * DIM;
  float* orow = out + (size_t)row * DIM;
  __shared__ float red[8];
  float s = 0.f;
  for (int i = threadIdx.x; i < DIM; i += 256) { float v = xr[i]; s += v * v; }
  for (int off = 16; off; off >>= 1) s += __shfl_xor(s, off, 32);
  int wv = threadIdx.x >> 5, ln = threadIdx.x & 31;
  if (ln == 0) red[wv] = s;
  __syncthreads();
  if (threadIdx.x == 0) { float t = 0.f; for (int i = 0; i < 8; ++i) t += red[i]; red[0] = t; }
  __syncthreads();
  float inv = rsqrtf(red[0] * (1.0f / DIM) + 1e-5f);
  for (int i = threadIdx.x; i < DIM; i += 256) orow[i] = xr[i] * inv * w[i];
}

// ---------------------------------------------------------------------------
// Skinny GEMM: C[0..7][N] += A16[16xK] @ W[KxN], WMMA f32 16x16x4.
// One wave per 16-wide N tile, split-K over blockIdx.y, atomic f32 accumulate.
// K unrolled by 8 (2 WMMAs/iter) + prefetch of the weight stream 32 rows ahead.
// ---------------------------------------------------------------------------
__global__ __launch_bounds__(128)
void gemm16_wmma(const float* __restrict__ A, const float* __restrict__ W,
                 float* __restrict__ C, int K, int N, int kChunk) {
  const int wavesPerBlk = blockDim.x >> 5;
  int wv   = threadIdx.x >> 5;
  int lane = threadIdx.x & 31;
  int tile = blockIdx.x * wavesPerBlk + wv;
  int n0   = tile * 16;
  int k0   = blockIdx.y * kChunk;
  int k1   = k0 + kChunk; if (k1 > K) k1 = K;
  int hf = lane >> 4;           // which half-wave
  int lr = lane & 15;
  const float* Ar = A + (size_t)lr * K;           // A row = M index
  v8f c = {0.f, 0.f, 0.f, 0.f, 0.f, 0.f, 0.f, 0.f};
  for (int k = k0; k < k1; k += 8) {
    // A 16x4: lanes0-15 -> K = k + {0,1}; lanes16-31 -> K = k+2 + {0,1}
    // B 4x16: VGPR0 rows k / k+2 (per half), VGPR1 rows k+1 / k+3; N contiguous
    v2f a0, b0, a1, b1;
    const float* Wp0 = W + (size_t)(k + 2 * hf) * N + n0 + lr;
    a0.x = Ar[k + 2 * hf];
    a0.y = Ar[k + 2 * hf + 1];
    b0.x = Wp0[0];
    b0.y = Wp0[N];
    const float* Wp1 = Wp0 + (size_t)4 * N;
    a1.x = Ar[k + 4 + 2 * hf];
    a1.y = Ar[k + 4 + 2 * hf + 1];
    b1.x = Wp1[0];
    b1.y = Wp1[N];
    __builtin_prefetch(Wp0 + (size_t)32 * N, 0, 0);  // global_prefetch_b8
    c = __builtin_amdgcn_wmma_f32_16x16x4_f32(false, a0, false, b0, (short)0, c,
                                              false, false);
    c = __builtin_amdgcn_wmma_f32_16x16x4_f32(false, a1, false, b1, (short)0, c,
                                              false, false);
  }
  if (lane < 16) {
    float* Cp = C + n0 + lane;                    // C VGPR i, lanes0-15 -> M=i
#pragma unroll
    for (int i = 0; i < 8; ++i) unsafeAtomicAdd(&Cp[(size_t)i * N], c[i]);
  }
}

// ---------------------------------------------------------------------------
// RoPE on new q,k (8 x 32 heads x 128), pos = start_pos
// ---------------------------------------------------------------------------
__global__ __launch_bounds__(256)
void rope_kernel(float* __restrict__ q, float* __restrict__ k,
                 const int* __restrict__ start_pos) {
  int idx  = blockIdx.x * 256 + threadIdx.x;   // 8*32*64 = 16384 pairs
  int pair = idx & 63;
  int bh   = idx >> 6;
  float pos = (float)(*start_pos);
  float inv_freq = __expf(-(float)(2 * pair) * (1.0f / HD) * 9.210340371976184f);
  float s, c;
  __sincosf(pos * inv_freq, &s, &c);
  int base = bh * HD + 2 * pair;
  float qr = q[base], qi = q[base + 1];
  q[base] = qr * c - qi * s;  q[base + 1] = qr * s + qi * c;
  float kr = k[base], ki = k[base + 1];
  k[base] = kr * c - ki * s;  k[base + 1] = kr * s + ki * c;
}

// ---------------------------------------------------------------------------
// Attention: one block per (b, head). Wave-per-key scores, softmax in LDS,
// then TDM double-buffered V streaming into LDS for the weighted-V pass.
// ---------------------------------------------------------------------------
__global__ __launch_bounds__(256)
void attn_kernel(const float* __restrict__ q, const float* __restrict__ knew,
                 const float* __restrict__ cache_k, const float* __restrict__ cache_v,
                 const float* __restrict__ vnew, const float* __restrict__ mask,
                 float* __restrict__ attn) {
  int b = blockIdx.x >> 5;
  int h = blockIdx.x & 31;
  __shared__ float sQ[HD];
  __shared__ float sS[SEQ];
  __shared__ float red[8];
  __shared__ float sV[2][VCH * HD];   // 2 x 16KB TDM double buffer
  int tid = threadIdx.x;
  if (tid < HD) sQ[tid] = q[(b * NH + h) * HD + tid];
  __syncthreads();

  int wv = tid >> 5, ln = tid & 31;
  const float scale = 0.08838834764831845f;  // 1/sqrt(128)
  for (int j = wv; j < SEQ; j += 8) {
    const float* kp = (j < SPN)
        ? (cache_k + (((size_t)b * SPN + j) * NH + h) * HD)
        : (knew + (size_t)(b * NH + h) * HD);
    float4 kv = ((const float4*)kp)[ln];
    float4 qv = ((const float4*)sQ)[ln];
    float d = kv.x * qv.x + kv.y * qv.y + kv.z * qv.z + kv.w * qv.w;
    for (int off = 16; off; off >>= 1) d += __shfl_xor(d, off, 32);
    if (ln == 0) sS[j] = d * scale + mask[((size_t)b * NH + h) * SEQ + j];
  }
  __syncthreads();

  // block max
  float m = -1e30f;
  for (int j = tid; j < SEQ; j += 256) m = fmaxf(m, sS[j]);
  for (int off = 16; off; off >>= 1) m = fmaxf(m, __shfl_xor(m, off, 32));
  if (ln == 0) red[wv] = m;
  __syncthreads();
  if (tid == 0) { float t = red[0]; for (int i = 1; i < 8; ++i) t = fmaxf(t, red[i]); red[0] = t; }
  __syncthreads();
  m = red[0];
  __syncthreads();

  // exp + block sum
  float s = 0.f;
  for (int j = tid; j < SEQ; j += 256) { float e = __expf(sS[j] - m); sS[j] = e; s += e; }
  for (int off = 16; off; off >>= 1) s += __shfl_xor(s, off, 32);
  if (ln == 0) red[wv] = s;
  __syncthreads();
  if (tid == 0) { float t = 0.f; for (int i = 0; i < 8; ++i) t += red[i]; red[0] = t; }
  __syncthreads();
  float denom = red[0];

  // ---- weighted V: TDM streams 32-row V chunks into LDS, double buffered ----
  const size_t vstride = (size_t)NH * HD;                       // row stride
  const float* vbase = cache_v + ((size_t)b * SPN * NH + h) * HD;
  if (wv == 0 && ln == 0) {                     // wave0 issues DMA chunk 0
    tdm_load_v((unsigned int)(uintptr_t)&sV[0][0], vbase, VCH);
  }
  if (wv == 0) __builtin_amdgcn_s_wait_tensorcnt(0);
  __syncthreads();

  float acc = 0.f;
  for (int c = 0; c < NCH; ++c) {
    if (wv == 0 && ln == 0 && (c + 1) < NCH) {  // prefetch next chunk via TDM
      tdm_load_v((unsigned int)(uintptr_t)&sV[(c + 1) & 1][0],
                 vbase + (size_t)(c + 1) * VCH * vstride, VCH);
    }
    if (tid < HD) {
      const float* vb_ = &sV[c & 1][0];
      int j0 = c * VCH;
#pragma unroll 8
      for (int r = 0; r < VCH; ++r) acc += sS[j0 + r] * vb_[r * HD + tid];
    }
    if (wv == 0) __builtin_amdgcn_s_wait_tensorcnt(0);
    __syncthreads();
  }
  if (tid < HD) {
    acc += sS[SPN] * vnew[(size_t)(b * NH + h) * HD + tid];
    attn[(size_t)b * DIM + h * HD + tid] = acc * (1.0f / denom);
  }
}

// ---------------------------------------------------------------------------
// SiLU(g1) * g3 -> gpad rows 0..7 (rows 8..15 stay zero)
// ---------------------------------------------------------------------------
__global__ __launch_bounds__(256)
void silumul_kernel(const float* __restrict__ g1, const float* __restrict__ g3,
                    float* __restrict__ gpad) {
  size_t i = (size_t)blockIdx.x * 256 + threadIdx.x;
  if (i < (size_t)NB * FFN) {
    float a  = g1[i];
    float sg = a / (1.f + __expf(-a));
    gpad[i]  = sg * g3[i];   // gpad rows 0..7 are first 8*FFN floats
  }
}

// ---------------------------------------------------------------------------
extern "C" void kernel_launch(void* const* d_in, const int* in_sizes, int n_in,
                              void* d_out, int out_size, void* d_ws, size_t ws_size,
                              hipStream_t stream) {
  const float* x           = (const float*)d_in[0];
  const float* cache_k     = (const float*)d_in[1];
  const float* cache_v     = (const float*)d_in[2];
  const float* mask        = (const float*)d_in[3];
  const float* wq          = (const float*)d_in[4];
  const float* wk          = (const float*)d_in[5];
  const float* wv          = (const float*)d_in[6];
  const float* wo          = (const float*)d_in[7];
  const float* w1          = (const float*)d_in[8];
  const float* w2          = (const float*)d_in[9];
  const float* w3          = (const float*)d_in[10];
  const float* attn_norm_w = (const float*)d_in[11];
  const float* ffn_norm_w  = (const float*)d_in[12];
  const int*   start_pos   = (const int*)d_in[13];
  float* out = (float*)d_out;

  float* ws = (float*)d_ws;
  size_t off = 0;
  auto alloc = [&](size_t n) { float* p = ws + off; off += n; return p; };
  float* hpad    = alloc((size_t)16 * DIM);   // rmsnorm1, 16-row padded
  float* qb      = alloc((size_t)NB * DIM);   // q,k,v contiguous
  float* kb      = alloc((size_t)NB * DIM);
  float* vb      = alloc((size_t)NB * DIM);
  float* attnpad = alloc((size_t)16 * DIM);   // attention out, padded
  float* x2      = alloc((size_t)NB * DIM);   // residual 1
  float* h2pad   = alloc((size_t)16 * DIM);   // rmsnorm2, padded
  float* g1      = alloc((size_t)NB * FFN);   // g1,g3 contiguous
  float* g3      = alloc((size_t)NB * FFN);
  float* gpad    = alloc((size_t)16 * FFN);   // silu*g3, padded

  // Re-zero everything accumulated or used as zero padding (graph-safe).
  (void)hipMemsetAsync(hpad,    0, (size_t)16 * DIM * sizeof(float), stream);
  (void)hipMemsetAsync(qb,      0, (size_t)3 * NB * DIM * sizeof(float), stream);
  (void)hipMemsetAsync(attnpad, 0, (size_t)16 * DIM * sizeof(float), stream);
  (void)hipMemsetAsync(h2pad,   0, (size_t)16 * DIM * sizeof(float), stream);
  (void)hipMemsetAsync(g1,      0, (size_t)2 * NB * FFN * sizeof(float), stream);
  (void)hipMemsetAsync(gpad,    0, (size_t)16 * FFN * sizeof(float), stream);

  // 1) RMSNorm
  rmsnorm_kernel<<<NB, 256, 0, stream>>>(x, attn_norm_w, hpad);

  // 2) QKV projections (K=4096, N=4096): 64 tiles-of-4-waves x 8 K-splits
  dim3 gq(DIM / 16 / 4, 8);
  gemm16_wmma<<<gq, 128, 0, stream>>>(hpad, wq, qb, DIM, DIM, 512);
  gemm16_wmma<<<gq, 128, 0, stream>>>(hpad, wk, kb, DIM, DIM, 512);
  gemm16_wmma<<<gq, 128, 0, stream>>>(hpad, wv, vb, DIM, DIM, 512);

  // 3) RoPE on q,k
  rope_kernel<<<(NB * NH * (HD / 2)) / 256, 256, 0, stream>>>(qb, kb, start_pos);

  // 4) Attention over 2049 keys (streams 536MB KV cache; V via TDM)
  attn_kernel<<<NB * NH, 256, 0, stream>>>(qb, kb, cache_k, cache_v, vb, mask,
                                           attnpad);

  // 5) x2 = x + attn @ wo  (accumulator pre-filled with x)
  (void)hipMemcpyAsync(x2, x, (size_t)NB * DIM * sizeof(float),
                       hipMemcpyDeviceToDevice, stream);
  gemm16_wmma<<<gq, 128, 0, stream>>>(attnpad, wo, x2, DIM, DIM, 512);

  // 6) RMSNorm 2
  rmsnorm_kernel<<<NB, 256, 0, stream>>>(x2, ffn_norm_w, h2pad);

  // 7) FFN up: g1 = h2 @ w1, g3 = h2 @ w3  (N=11008 -> 172 tile-blocks)
  dim3 gf(FFN / 16 / 4, 8);
  gemm16_wmma<<<gf, 128, 0, stream>>>(h2pad, w1, g1, DIM, FFN, 512);
  gemm16_wmma<<<gf, 128, 0, stream>>>(h2pad, w3, g3, DIM, FFN, 512);

  // 8) g = silu(g1) * g3
  silumul_kernel<<<((NB * FFN) + 255) / 256, 256, 0, stream>>>(g1, g3, gpad);

  // 9) out = x2 + g @ w2  (K=11008 -> 22 K-splits, last chunk 256)
  (void)hipMemcpyAsync(out, x2, (size_t)NB * DIM * sizeof(float),
                       hipMemcpyDeviceToDevice, stream);
  dim3 gd(DIM / 16 / 4, (FFN + 511) / 512);
  gemm16_wmma<<<gd, 128, 0, stream>>>(gpad, w2, out, FFN, DIM, 512);
}